// MapLayer_71914932404445
// MI455X (gfx1250) — compile-verified
//
#include <hip/hip_runtime.h>
#include <hip/hip_bf16.h>
#include <cstdint>

#define NDET     8400
#define SORT_N   16384
#define NCLS     80
#define NMASK    32
#define MHW      160
#define PROTO_PX (MHW * MHW)   // 25600
#define HW       640
#define KTOP     100
#define CONF_T   0.4f
#define IOU_T    0.7f

typedef __attribute__((ext_vector_type(2))) float        v2f;
typedef __attribute__((ext_vector_type(8))) float        v8f;
typedef __attribute__((ext_vector_type(4))) unsigned int v4u;
typedef __attribute__((ext_vector_type(8))) int          v8i;
typedef __attribute__((ext_vector_type(4))) int          v4i;

// ---------------------------------------------------------------------------
// Kernel A: per-anchor class max/argmax + stable sort key construction.
// Key = (order-preserving-u32(valid ? -score : +inf) << 32) | anchor_index
// so ascending u64 sort == argsort(where(valid,-s,inf)) with stable ties.
// ---------------------------------------------------------------------------
__global__ __launch_bounds__(256) void score_kernel(const float* __restrict__ det,
                                                    unsigned long long* __restrict__ keys,
                                                    float* __restrict__ score,
                                                    int* __restrict__ cid) {
  int n = blockIdx.x * 256 + threadIdx.x;
  if (n >= SORT_N) return;
  if (n >= NDET) { keys[n] = ~0ull; return; }   // pad: sorts after everything
  float best = -1e30f; int bi = 0;
  #pragma unroll 4
  for (int c = 0; c < NCLS; ++c) {
    float v = det[(4 + c) * NDET + n];
    if (v > best) { best = v; bi = c; }         // strict '>' => first argmax
  }
  score[n] = best;
  cid[n]   = bi;
  float kf = (best >= CONF_T) ? -best : __builtin_huge_valf();
  unsigned u = __float_as_uint(kf);
  u = u ^ ((unsigned)((int)u >> 31) | 0x80000000u);   // monotonic float->uint
  keys[n] = ((unsigned long long)u << 32) | (unsigned)n;
}

// ---------------------------------------------------------------------------
// Kernel B: single-workgroup bitonic sort of 16384 u64 keys in 128 KB LDS
// (fits because CDNA5 has 320 KB LDS per workgroup). Emits order[pos].
// ---------------------------------------------------------------------------
__global__ __launch_bounds__(1024) void sort_kernel(const unsigned long long* __restrict__ keys,
                                                    int* __restrict__ order) {
  extern __shared__ unsigned long long sk[];
  const int tid = threadIdx.x;
  for (int i = tid; i < SORT_N; i += 1024) sk[i] = keys[i];
  __syncthreads();
  for (int k = 2; k <= SORT_N; k <<= 1) {
    for (int j = k >> 1; j > 0; j >>= 1) {
      for (int i = tid; i < SORT_N; i += 1024) {
        int ixj = i ^ j;
        if (ixj > i) {
          unsigned long long a = sk[i], b = sk[ixj];
          bool up = (i & k) == 0;
          if ((a > b) == up) { sk[i] = b; sk[ixj] = a; }
        }
      }
      __syncthreads();
    }
  }
  for (int i = tid; i < NDET; i += 1024)
    order[i] = (int)(unsigned)(sk[i] & 0xFFFFFFFFull);
}

// ---------------------------------------------------------------------------
// Kernel C: greedy NMS, single workgroup. Boxes live in LDS (~140 KB).
// Uniform forward scan to next kept entry (broadcast LDS reads, no barrier),
// one parallel suppression sweep + barrier per kept box, early exit at KTOP
// kept (the first KTOP kept in sorted order are exactly lax.top_k's picks).
// ---------------------------------------------------------------------------
__global__ __launch_bounds__(1024) void nms_kernel(const float* __restrict__ det,
                                                   const float* __restrict__ score,
                                                   const int* __restrict__ cid,
                                                   const int* __restrict__ order,
                                                   int* __restrict__ topDet,
                                                   int* __restrict__ topBox,
                                                   int* __restrict__ topCls,
                                                   int* __restrict__ topValid,
                                                   int* __restrict__ keptCount) {
  extern __shared__ unsigned char nms_smem[];
  float* bx1 = (float*)nms_smem;
  float* by1 = bx1 + NDET;
  float* bx2 = by1 + NDET;
  float* by2 = bx2 + NDET;
  unsigned char* keep = (unsigned char*)(by2 + NDET);
  const int tid = threadIdx.x;

  for (int i = tid; i < NDET; i += 1024) {
    int oi = order[i];
    bx1[i] = det[0 * NDET + oi];
    by1[i] = det[1 * NDET + oi];
    bx2[i] = det[2 * NDET + oi];
    by2[i] = det[3 * NDET + oi];
    keep[i] = (score[oi] >= CONF_T) ? 1 : 0;
  }
  __syncthreads();

  int count = 0;
  int i = 0;
  while (i < NDET && count < KTOP) {
    while (i < NDET && keep[i] == 0) ++i;   // uniform: all lanes read same LDS byte
    if (i >= NDET) break;
    float x1 = bx1[i], y1 = by1[i], x2 = bx2[i], y2 = by2[i];
    float ai = (x2 - x1) * (y2 - y1);
    if (tid == 0) {
      int oi = order[i];
      topDet[count]       = oi;
      topBox[count*4 + 0] = (int)x1;   // trunc-toward-zero == astype(int32), coords >= 0
      topBox[count*4 + 1] = (int)y1;
      topBox[count*4 + 2] = (int)x2;
      topBox[count*4 + 3] = (int)y2;
      topCls[count]       = cid[oi];
      topValid[count]     = 1;
    }
    for (int j = i + 1 + tid; j < NDET; j += 1024) {
      if (!keep[j]) continue;
      float ix1 = fmaxf(x1, bx1[j]);
      float iy1 = fmaxf(y1, by1[j]);
      float ix2 = fminf(x2, bx2[j]);
      float iy2 = fminf(y2, by2[j]);
      float inter = fmaxf(ix2 - ix1, 0.0f) * fmaxf(iy2 - iy1, 0.0f);
      float aj = (bx2[j] - bx1[j]) * (by2[j] - by1[j]);
      float iou = inter / (ai + aj - inter);
      if (iou >= IOU_T) keep[j] = 0;
    }
    ++count;
    ++i;
    __syncthreads();
  }
  if (tid == 0) *keptCount = count;
  for (int k = count + tid; k < KTOP; k += 1024) {   // padding slots: never painted
    topValid[k]   = 0;
    topDet[k]     = order[0];
    topCls[k]     = 0;
    topBox[k*4+0] = 0; topBox[k*4+1] = 0; topBox[k*4+2] = 0; topBox[k*4+3] = 0;
  }
}

// ---------------------------------------------------------------------------
// Kernel D: mask logits GEMM via V_WMMA_F32_16X16X4_F32, with the proto
// B-panel (32 rows x 16 cols, row stride 25600 f32) DMA'd into LDS by the
// Tensor Data Mover (one tensor_load_to_lds per wave, completion via
// s_wait_tensorcnt). B is read back with explicit ds_load_b32 inline asm
// (one block: 16 loads + s_wait_dscnt 0) so the lowering is exactly the DS
// path -- neither folded away (round 2) nor demoted to serialized volatile
// flat loads (round 3). hard mask = logit > 0 (== sigmoid > 0.5).
// A f32 16x4 layout: lanes 0-15 row M=lane, VGPR0 K={0|2}, VGPR1 K={1|3}.
// C/D f32 16x16 layout: VGPR r -> M = r + (lane<16 ? 0 : 8), N = lane%16.
// ---------------------------------------------------------------------------
__global__ __launch_bounds__(128) void mask_wmma_kernel(const float* __restrict__ det,
                                                        const float* __restrict__ proto,
                                                        const int* __restrict__ topDet,
                                                        unsigned char* __restrict__ masks) {
  // 4 waves x (32x16) f32 tiles (bytes 0..8191, written only by the TDM)
  // + 32-float pad that we store to, keeping the LDS block allocated.
  __shared__ float btile[4 * NMASK * 16 + 32];
  const int wave = __builtin_amdgcn_readfirstlane((int)(threadIdx.x >> 5));
  const int lane = (int)(threadIdx.x & 31u);
  const int ct   = blockIdx.x * 4 + wave;  // column tile: 0..1599 (25600/16)
  const int mt   = blockIdx.y;             // row tile:    0..6    (112 >= KTOP)
  const int pBase   = ct * 16;
  const int rowBase = mt * 16;
  const int half = lane >> 4;              // 0: lanes 0-15, 1: lanes 16-31
  const int l16  = lane & 15;

  // Keep btile alive (store lands in the pad region, outside the DMA target;
  // survives DSE because the asm block below clobbers memory).
  if (threadIdx.x < 32) btile[4 * NMASK * 16 + threadIdx.x] = 0.0f;

  // ---- TDM descriptor: 2D tile (tile_dim0=16 f32 wide, tile_dim1=32 rows)
  //      out of tensor (dim0=25600, dim0_stride=25600), into this wave's LDS slot.
  {
    unsigned long long gaddr = (unsigned long long)(uintptr_t)(proto + pBase);
    unsigned int lds_off = (unsigned int)(wave * (NMASK * 16 * 4)); // btile @ LDS offset 0
    v4u g0 = { 1u,                                            // [1:0] count=1
               lds_off,                                       // [63:32] lds_addr
               (unsigned int)(gaddr & 0xFFFFFFFFull),         // [95:64] global_addr lo
               (unsigned int)((gaddr >> 32) & 0x01FFFFFFull)  // [120:96] global_addr hi
                 | (2u << 30) };                              // [127:126] type=2 (image)
    v8i g1 = { (int)0x00020000u,            // [17:16] data_size=2 (4B), wg_mask=0
               (int)0x64000000u,            // [63:48] tensor_dim0 lo16 = 25600
               (int)(32u << 16),            // [95:80] tensor_dim1 lo16 = 32
               (int)(16u << 16),            // [127:112] tile_dim0 = 16
               (int)32,                     // [143:128] tile_dim1 = 32 (tile_dim2 = 0)
               (int)PROTO_PX,               // [191:160] tensor_dim0_stride lo32 = 25600
               0, 0 };                      // dim0_stride hi / dim1_stride = 0
    v4i gz = { 0, 0, 0, 0 };
#if __has_include(<hip/amd_detail/amd_gfx1250_TDM.h>)
    v8i gz8 = { 0, 0, 0, 0, 0, 0, 0, 0 };   // therock/clang-23: 6-arg builtin
    __builtin_amdgcn_tensor_load_to_lds(g0, g1, gz, gz, gz8, 0);
#else
    __builtin_amdgcn_tensor_load_to_lds(g0, g1, gz, gz, 0);  // ROCm 7.2: 5-arg builtin
#endif
  }

  // ---- A operand gathers (overlap with the TDM transfer)
  int slot  = rowBase + l16;
  int slotc = (slot < KTOP) ? slot : (KTOP - 1);
  int oi    = topDet[slotc];
  const float* coeff = det + (4 + NCLS) * NDET + oi;  // coeff[c] at c*NDET
  float areg[16];
  #pragma unroll
  for (int kb = 0; kb < 8; ++kb) {
    const int k0 = kb * 4 + half * 2;
    areg[kb * 2 + 0] = coeff[(k0 + 0) * NDET];
    areg[kb * 2 + 1] = coeff[(k0 + 1) * NDET];
  }

  // CDNA5 global_prefetch_b8: warm L2 for the next block's proto tiles
  if (ct + 4 < PROTO_PX / 16) __builtin_prefetch(proto + pBase + 64, 0, 1);

  // ---- wait for the tensor DMA, then pull the B panel out of LDS with
  // explicit ds_load_b32 (element (row r, col l16) of this wave's tile sits
  // at byte wave*2048 + r*64 + l16*4; the per-lane part goes in the address
  // VGPR, the row stride in the immediate offset field).
  __builtin_amdgcn_s_wait_tensorcnt(0);
  int laddr = wave * 2048 + l16 * 4 + half * 128;
  float b0, b1, b2, b3, b4, b5, b6, b7, b8, b9, b10, b11, b12, b13, b14, b15;
  asm volatile(
      "ds_load_b32 %0,  %16 offset:0\n\t"      // kb=0 rows k0,k0+1
      "ds_load_b32 %1,  %16 offset:64\n\t"
      "ds_load_b32 %2,  %16 offset:256\n\t"    // kb=1
      "ds_load_b32 %3,  %16 offset:320\n\t"
      "ds_load_b32 %4,  %16 offset:512\n\t"    // kb=2
      "ds_load_b32 %5,  %16 offset:576\n\t"
      "ds_load_b32 %6,  %16 offset:768\n\t"    // kb=3
      "ds_load_b32 %7,  %16 offset:832\n\t"
      "ds_load_b32 %8,  %16 offset:1024\n\t"   // kb=4
      "ds_load_b32 %9,  %16 offset:1088\n\t"
      "ds_load_b32 %10, %16 offset:1280\n\t"   // kb=5
      "ds_load_b32 %11, %16 offset:1344\n\t"
      "ds_load_b32 %12, %16 offset:1536\n\t"   // kb=6
      "ds_load_b32 %13, %16 offset:1600\n\t"
      "ds_load_b32 %14, %16 offset:1792\n\t"   // kb=7
      "ds_load_b32 %15, %16 offset:1856\n\t"
      "s_wait_dscnt 0x0"
      : "=&v"(b0), "=&v"(b1), "=&v"(b2), "=&v"(b3),
        "=&v"(b4), "=&v"(b5), "=&v"(b6), "=&v"(b7),
        "=&v"(b8), "=&v"(b9), "=&v"(b10), "=&v"(b11),
        "=&v"(b12), "=&v"(b13), "=&v"(b14), "=&v"(b15)
      : "v"(laddr)
      : "memory");
  float bv[16] = {b0, b1, b2, b3, b4, b5, b6, b7,
                  b8, b9, b10, b11, b12, b13, b14, b15};

  v8f acc = {0.f, 0.f, 0.f, 0.f, 0.f, 0.f, 0.f, 0.f};
  #pragma unroll
  for (int kb = 0; kb < 8; ++kb) {
    v2f a, b;
    a.x = areg[kb * 2 + 0];
    a.y = areg[kb * 2 + 1];
    b.x = bv[kb * 2 + 0];
    b.y = bv[kb * 2 + 1];
    acc = __builtin_amdgcn_wmma_f32_16x16x4_f32(false, a, false, b,
                                                (short)0, acc, false, false);
  }

  const int p = pBase + l16;
  #pragma unroll
  for (int r = 0; r < 8; ++r) {
    int s = rowBase + r + half * 8;
    if (s < KTOP) masks[(size_t)s * PROTO_PX + p] = (acc[r] > 0.0f) ? 1 : 0;
  }
}

// ---------------------------------------------------------------------------
// Kernel E: paint canvas (640,640,80). Last covering valid box wins; mask
// upsample(4x bilinear, half-pixel) > 0  ==  OR of the <=4 in-range neighbor
// texels (all bilinear weights here are in {0.125,0.375,0.625,0.875}).
// ---------------------------------------------------------------------------
__global__ __launch_bounds__(256) void paint_kernel(const int* __restrict__ topBox,
                                                    const int* __restrict__ topCls,
                                                    const int* __restrict__ topValid,
                                                    const unsigned char* __restrict__ masks,
                                                    float* __restrict__ out) {
  __shared__ int s_b[KTOP * 4];
  __shared__ int s_c[KTOP];
  const int tid = threadIdx.x;
  if (tid < KTOP) {
    int v  = topValid[tid];
    int x1 = topBox[tid*4+0], y1 = topBox[tid*4+1];
    int x2 = topBox[tid*4+2], y2 = topBox[tid*4+3];
    if (!v) { x1 = 0; y1 = 0; x2 = 0; y2 = 0; }   // empty box: never covers
    s_b[tid*4+0] = x1; s_b[tid*4+1] = y1; s_b[tid*4+2] = x2; s_b[tid*4+3] = y2;
    s_c[tid] = topCls[tid];
  }
  __syncthreads();

  const int pid = blockIdx.x * 256 + tid;          // 0..409599
  const int y = pid / HW, x = pid % HW;

  int cover = -1;
  #pragma unroll 4
  for (int k = KTOP - 1; k >= 0; --k) {
    if (x >= s_b[k*4+0] && x < s_b[k*4+2] && y >= s_b[k*4+1] && y < s_b[k*4+3]) {
      cover = k;
      break;
    }
  }

  float4* o = (float4*)(out + (size_t)pid * NCLS);
  if (cover < 0) {
    float4 z = {0.f, 0.f, 0.f, 0.f};
    #pragma unroll
    for (int g = 0; g < NCLS / 4; ++g) o[g] = z;
    return;
  }

  float sy = y * 0.25f - 0.375f;
  float sx = x * 0.25f - 0.375f;
  int iy0 = (int)floorf(sy), ix0 = (int)floorf(sx);
  int cy0 = iy0 < 0 ? 0 : iy0;
  int cy1 = (iy0 + 1 > MHW - 1) ? MHW - 1 : iy0 + 1;
  int cx0 = ix0 < 0 ? 0 : ix0;
  int cx1 = (ix0 + 1 > MHW - 1) ? MHW - 1 : ix0 + 1;
  const unsigned char* m = masks + (size_t)cover * PROTO_PX;
  int bit = m[cy0*MHW + cx0] | m[cy0*MHW + cx1] | m[cy1*MHW + cx0] | m[cy1*MHW + cx1];
  float val = bit ? 1.0f : 0.0f;
  int cls = s_c[cover];
  #pragma unroll
  for (int g = 0; g < NCLS / 4; ++g) {
    float4 v = {0.f, 0.f, 0.f, 0.f};
    int base = g * 4;
    if (cls >= base && cls < base + 4) ((float*)&v)[cls - base] = val;
    o[g] = v;
  }
}

// ---------------------------------------------------------------------------
extern "C" void kernel_launch(void* const* d_in, const int* in_sizes, int n_in,
                              void* d_out, int out_size, void* d_ws, size_t ws_size,
                              hipStream_t stream) {
  const float* det   = (const float*)d_in[0];   // (1,116,8400) f32
  const float* proto = (const float*)d_in[1];   // (1,32,160,160) f32
  float* out = (float*)d_out;                   // (1,640,640,80) f32

  // Workspace layout (bytes)
  uint8_t* ws = (uint8_t*)d_ws;
  unsigned long long* keys = (unsigned long long*)(ws + 0);        // 131072
  float* score  = (float*)(ws + 131072);                           // 33600
  int*   cid    = (int*)  (ws + 164672);                           // 33600
  int*   order  = (int*)  (ws + 198272);                           // 33600
  int*   topDet = (int*)  (ws + 231872);                           // 400
  int*   topBox = (int*)  (ws + 232272);                           // 1600
  int*   topCls = (int*)  (ws + 233872);                           // 400
  int*   topVal = (int*)  (ws + 234272);                           // 400
  int*   keptCt = (int*)  (ws + 234672);                           // 4
  unsigned char* masks = (unsigned char*)(ws + 234752);            // 2,560,000

  // Allow >64KB dynamic LDS on the single-workgroup kernels (CDNA5: 320KB/WGP)
  (void)hipFuncSetAttribute((const void*)sort_kernel,
                            hipFuncAttributeMaxDynamicSharedMemorySize,
                            SORT_N * (int)sizeof(unsigned long long));
  (void)hipFuncSetAttribute((const void*)nms_kernel,
                            hipFuncAttributeMaxDynamicSharedMemorySize,
                            NDET * 4 * (int)sizeof(float) + NDET);

  score_kernel<<<SORT_N / 256, 256, 0, stream>>>(det, keys, score, cid);
  sort_kernel<<<1, 1024, SORT_N * sizeof(unsigned long long), stream>>>(keys, order);
  nms_kernel<<<1, 1024, NDET * 4 * sizeof(float) + NDET, stream>>>(
      det, score, cid, order, topDet, topBox, topCls, topVal, keptCt);
  mask_wmma_kernel<<<dim3(PROTO_PX / 16 / 4, 7, 1), 128, 0, stream>>>(
      det, proto, topDet, masks);
  paint_kernel<<<(HW * HW) / 256, 256, 0, stream>>>(topBox, topCls, topVal, masks, out);
}